// HybridGNN_68255620268692
// MI455X (gfx1250) — compile-verified
//
#include <hip/hip_runtime.h>
#include <hip/hip_bf16.h>
#include <math.h>

typedef float v2f __attribute__((ext_vector_type(2)));
typedef float v8f __attribute__((ext_vector_type(8)));

#define B_   8
#define W_   20
#define M_   2048
#define HSC_ 16
#define HID_ 32
#define NH_  32
#define DG_  80

__device__ __forceinline__ float sigmoidf_(float x){ return 1.0f/(1.0f+__expf(-x)); }
__device__ __forceinline__ float eluf_(float x){ return x > 0.0f ? x : (__expf(x)-1.0f); }

// ---------------------------------------------------------------------------
// 1) h_SC = x^T Wt + bt; q = h_SC WQ + bq; k = h_SC WK + bk   (per (b,m))
// ---------------------------------------------------------------------------
__global__ void k_timeconv(const float* __restrict__ x, const float* __restrict__ Wt,
                           const float* __restrict__ bt, const float* __restrict__ WQw,
                           const float* __restrict__ WQb, const float* __restrict__ WKw,
                           const float* __restrict__ WKb,
                           float* __restrict__ hsc, float* __restrict__ q, float* __restrict__ kk)
{
    __shared__ float sWt[W_*HSC_], sbt[HSC_], sQw[HSC_*HID_], sQb[HID_], sKw[HSC_*HID_], sKb[HID_];
    int tid = threadIdx.x;
    for (int i = tid; i < W_*HSC_; i += 256) sWt[i] = Wt[i];
    if (tid < HSC_) sbt[tid] = bt[tid];
    for (int i = tid; i < HSC_*HID_; i += 256) { sQw[i] = WQw[i]; sKw[i] = WKw[i]; }
    if (tid < HID_) { sQb[tid] = WQb[tid]; sKb[tid] = WKb[tid]; }
    __syncthreads();
    int gid = blockIdx.x*256 + tid;          // b*M + m
    int b = gid / M_, m = gid % M_;
    float xv[W_];
    #pragma unroll
    for (int t = 0; t < W_; ++t) xv[t] = x[(size_t)(b*W_ + t)*M_ + m];
    float h[HSC_];
    #pragma unroll
    for (int k = 0; k < HSC_; ++k) {
        float a = sbt[k];
        #pragma unroll
        for (int t = 0; t < W_; ++t) a += xv[t]*sWt[t*HSC_+k];
        h[k] = a;
        hsc[(size_t)gid*HSC_ + k] = a;
    }
    #pragma unroll
    for (int j = 0; j < HID_; ++j) {
        float aq = sQb[j], ak = sKb[j];
        #pragma unroll
        for (int k2 = 0; k2 < HSC_; ++k2) { aq += h[k2]*sQw[k2*HID_+j]; ak += h[k2]*sKw[k2*HID_+j]; }
        q [(size_t)gid*HID_ + j] = aq;
        kk[(size_t)gid*HID_ + j] = ak;
    }
}

// ---------------------------------------------------------------------------
// 2) per-batch Gram G = K^T K  (32x32)  and ksum = sum_m k
// ---------------------------------------------------------------------------
__global__ void k_gram(const float* __restrict__ kk, float* __restrict__ G, float* __restrict__ ksum)
{
    __shared__ float KL[32][33];
    int b = blockIdx.x;
    int tid = threadIdx.x;                   // 1024 threads
    int h1 = tid >> 5, h2 = tid & 31;
    float g = 0.f, ks = 0.f;
    for (int m0 = 0; m0 < M_; m0 += 32) {
        KL[tid >> 5][tid & 31] = kk[(size_t)(b*M_ + m0 + (tid >> 5))*HID_ + (tid & 31)];
        __syncthreads();
        #pragma unroll
        for (int mm = 0; mm < 32; ++mm) g += KL[mm][h1]*KL[mm][h2];
        if (tid < 32) {
            #pragma unroll
            for (int mm = 0; mm < 32; ++mm) ks += KL[mm][tid];
        }
        __syncthreads();
    }
    G[b*1024 + tid] = g;
    if (tid < 32) ksum[b*32 + tid] = ks;
}

// ---------------------------------------------------------------------------
// 3) node0 = [h_SC | h_L | h_G]  using S1=q.ksum, S2=q^T G q (closed-form attn)
// ---------------------------------------------------------------------------
__global__ void k_node0(const float* __restrict__ hsc, const float* __restrict__ q,
                        const float* __restrict__ G, const float* __restrict__ ksum,
                        const float* __restrict__ tencw, const float* __restrict__ tencb,
                        const float* __restrict__ sencw, const float* __restrict__ sencb,
                        const float* __restrict__ degree, float* __restrict__ node0)
{
    __shared__ float sG[1024], sks[32], stw[32], stb[32], ssw[32], ssb[32];
    int tid = threadIdx.x;
    int b = blockIdx.x >> 3;                 // 8 blocks per batch
    for (int i = tid; i < 1024; i += 256) sG[i] = G[b*1024 + i];
    if (tid < 32) { sks[tid]=ksum[b*32+tid]; stw[tid]=tencw[tid]; stb[tid]=tencb[tid];
                    ssw[tid]=sencw[tid]; ssb[tid]=sencb[tid]; }
    __syncthreads();
    int gid = blockIdx.x*256 + tid;          // b*M + m
    int m = gid % M_;
    float qv[HID_];
    #pragma unroll
    for (int j = 0; j < HID_; ++j) qv[j] = q[(size_t)gid*HID_ + j];
    float S1 = 0.f, S2 = 0.f;
    #pragma unroll
    for (int a = 0; a < HID_; ++a) {
        S1 += qv[a]*sks[a];
        float t = 0.f;
        #pragma unroll
        for (int c = 0; c < HID_; ++c) t += sG[a*32+c]*qv[c];
        S2 += qv[a]*t;
    }
    float rn = S1 / fmaxf(sqrtf(fmaxf(S2, 0.f)), 1e-12f);
    float dg = degree[m];
    float* np = node0 + (size_t)gid*DG_;
    #pragma unroll
    for (int k2 = 0; k2 < HSC_; ++k2) np[k2] = hsc[(size_t)gid*HSC_+k2];
    #pragma unroll
    for (int j = 0; j < HID_; ++j) np[HSC_+j] = dg*ssw[j] + ssb[j];
    #pragma unroll
    for (int j = 0; j < HID_; ++j) np[HSC_+HID_+j] = rn*stw[j] + stb[j];
}

// ---------------------------------------------------------------------------
// 4) GRU (20 steps, hidden 32) per sequence + ann score
// ---------------------------------------------------------------------------
__global__ void k_gru(const float* __restrict__ x, const float* __restrict__ wih,
                      const float* __restrict__ whh, const float* __restrict__ bih,
                      const float* __restrict__ bhh, const float* __restrict__ W1,
                      const float* __restrict__ b1p, const float* __restrict__ W2,
                      const float* __restrict__ Vv, const float* __restrict__ bv,
                      float* __restrict__ hT, float* __restrict__ ann)
{
    __shared__ float sW[96*32], swih[96], sbih[96], sbhh[96], sw12[16*32], sb1[16], sV[16];
    int tid = threadIdx.x;
    for (int i = tid; i < 96*32; i += 256) sW[i] = whh[i];
    if (tid < 96) { swih[tid]=wih[tid]; sbih[tid]=bih[tid]; sbhh[tid]=bhh[tid]; }
    for (int i = tid; i < 16*32; i += 256) sw12[i] = W1[i] + W2[i];
    if (tid < 16) { sb1[tid]=b1p[tid]; sV[tid]=Vv[tid]; }
    __syncthreads();
    int gid = blockIdx.x*256 + tid;
    int b = gid / M_, m = gid % M_;
    float h[32];
    #pragma unroll
    for (int i = 0; i < 32; ++i) h[i] = 0.f;
    for (int t = 0; t < W_; ++t) {
        float xt = x[(size_t)(b*W_+t)*M_ + m];
        float hn2[32];
        #pragma unroll
        for (int i = 0; i < 32; ++i) {
            float hr = sbhh[i], hz = sbhh[32+i], hg = sbhh[64+i];
            #pragma unroll
            for (int j = 0; j < 32; ++j) {
                hr += sW[i*32+j]*h[j];
                hz += sW[(32+i)*32+j]*h[j];
                hg += sW[(64+i)*32+j]*h[j];
            }
            float r = sigmoidf_(xt*swih[i]    + sbih[i]    + hr);
            float z = sigmoidf_(xt*swih[32+i] + sbih[32+i] + hz);
            float n = tanhf    (xt*swih[64+i] + sbih[64+i] + r*hg);
            hn2[i] = (1.f - z)*n + z*h[i];
        }
        #pragma unroll
        for (int i = 0; i < 32; ++i) h[i] = hn2[i];
    }
    #pragma unroll
    for (int i = 0; i < 32; ++i) hT[(size_t)gid*32 + i] = h[i];
    float s = bv[0];
    #pragma unroll
    for (int i = 0; i < 16; ++i) {
        float a = sb1[i];
        #pragma unroll
        for (int j = 0; j < 32; ++j) a += h[j]*sw12[i*32+j];
        s += eluf_(a)*sV[i];
    }
    ann[gid] = s / fmaxf(sqrtf((float)M_)*fabsf(s), 1e-12f);
}

// ---------------------------------------------------------------------------
// 5) c_gate pre-activation = ann @ Wb, deterministic split-K partials
// ---------------------------------------------------------------------------
__global__ void k_cgate_part(const float* __restrict__ ann, const float* __restrict__ Wb,
                             float* __restrict__ part)
{
    __shared__ float sA[8][128];
    int tid = threadIdx.x;                   // 128 threads
    int j  = blockIdx.x*128 + tid;
    int k0 = blockIdx.y*128;
    for (int b = 0; b < 8; ++b) sA[b][tid] = ann[b*M_ + k0 + tid];
    __syncthreads();
    float acc[8] = {};
    for (int kk = 0; kk < 128; ++kk) {
        float w = Wb[(size_t)(k0+kk)*M_ + j];
        #pragma unroll
        for (int b = 0; b < 8; ++b) acc[b] += sA[b][kk]*w;
    }
    #pragma unroll
    for (int b = 0; b < 8; ++b) part[((size_t)blockIdx.y*8 + b)*M_ + j] = acc[b];
}

__global__ void k_cgate_fin(const float* __restrict__ part, const float* __restrict__ ann,
                            const float* __restrict__ wb, float* __restrict__ cg,
                            float* __restrict__ colw)
{
    int gid = blockIdx.x*256 + threadIdx.x;  // b*M + j
    int b = gid / M_, j = gid % M_;
    float a = wb[0];
    #pragma unroll
    for (int kc = 0; kc < 16; ++kc) a += part[((size_t)kc*8 + b)*M_ + j];
    float c = sigmoidf_(a);
    cg[gid]   = c;
    colw[gid] = ann[gid]*(1.f - c);
}

// ---------------------------------------------------------------------------
// 6) spatial = sigmoid(d_gate * deg_i * deg_j) * adj_orig   (batch-invariant)
// ---------------------------------------------------------------------------
__global__ void k_spatial(const float* __restrict__ dgate, const float* __restrict__ adjo,
                          const float* __restrict__ degree, float* __restrict__ spat)
{
    size_t idx = (size_t)blockIdx.x*256 + threadIdx.x;
    int i = (int)(idx / M_), j = (int)(idx % M_);
    spat[idx] = sigmoidf_(dgate[idx]*degree[i]*degree[j]) * adjo[idx];
}

// ---------------------------------------------------------------------------
// 7) inv_deg[b,i] = 1 / #(adj_fin[b,i,:] > 0)
// ---------------------------------------------------------------------------
__global__ void k_deg(const float* __restrict__ adjg, const float* __restrict__ spat,
                      const float* __restrict__ cg, const float* __restrict__ colw,
                      float* __restrict__ invdeg)
{
    __shared__ float red[256];
    int bi = blockIdx.x;                     // b*M + i
    int b = bi / M_, i = bi % M_;
    int tid = threadIdx.x;
    const float* ga  = adjg + (size_t)i*M_;
    const float* sp  = spat + (size_t)i*M_;
    const float* cgb = cg   + (size_t)b*M_;
    const float* cwb = colw + (size_t)b*M_;
    float cnt = 0.f;
    for (int j = tid; j < M_; j += 256) {
        float v = ga[j]*cgb[j] + cwb[j] + sp[j];
        cnt += (v > 0.f) ? 1.f : 0.f;
    }
    red[tid] = cnt;
    __syncthreads();
    for (int s = 128; s > 0; s >>= 1) { if (tid < s) red[tid] += red[tid+s]; __syncthreads(); }
    if (tid == 0) invdeg[bi] = red[0] > 0.f ? 1.f/red[0] : 0.f;
}

// ---------------------------------------------------------------------------
// per-layer helpers: column-scaled node, rank-1 term
// ---------------------------------------------------------------------------
__global__ void k_nodeA(const float* __restrict__ node, const float* __restrict__ cg,
                        float* __restrict__ nodeA)
{
    size_t idx = (size_t)blockIdx.x*256 + threadIdx.x;
    size_t bm = idx / DG_;
    nodeA[idx] = cg[bm]*node[idx];
}

__global__ void k_tvec(const float* __restrict__ node, const float* __restrict__ colw,
                       float* __restrict__ tvec)
{
    int b = blockIdx.x, d = threadIdx.x;     // 80 threads
    float a = 0.f;
    for (int j = 0; j < M_; ++j) a += colw[b*M_+j]*node[((size_t)(b*M_+j))*DG_ + d];
    tvec[b*DG_ + d] = a;
}

// ---------------------------------------------------------------------------
// 8) WMMA f32 16x16x4: agg = inv_deg * (adj_geo @ nodeA + spatial @ node + t)
//    LDS B tiles stored k-pair interleaved: row pair (2p,2p+1), col d at
//    [p*160 + 2d + {0,1}] -> each B fragment is ONE aligned ds_load_b64.
// ---------------------------------------------------------------------------
__global__ void __launch_bounds__(256) k_aggmm(
    const float* __restrict__ adjg, const float* __restrict__ spat,
    const float* __restrict__ node, const float* __restrict__ nodeA,
    const float* __restrict__ tvec, const float* __restrict__ invdeg,
    float* __restrict__ agg)
{
    __shared__ float sA[32*80];              // cg-scaled node chunk (interleaved)
    __shared__ float sN[32*80];              // plain node chunk     (interleaved)
    int tid  = threadIdx.x;
    int lane = tid & 31, wave = tid >> 5;
    int hlf  = lane >> 4, ln = lane & 15;
    int b  = blockIdx.y;
    int i0 = blockIdx.x*128 + wave*16;
    int rowA = i0 + ln;
    const float* gaRow = adjg + (size_t)rowA*M_;
    const float* spRow = spat + (size_t)rowA*M_;
    v8f acc[5] = {};
    for (int k0 = 0; k0 < M_; k0 += 32) {
        __syncthreads();
        for (int idx = tid; idx < 32*80; idx += 256) {
            int kk2 = idx / 80, d = idx % 80;
            size_t g = ((size_t)(b*M_ + k0 + kk2))*DG_ + d;
            int li = (kk2 >> 1)*160 + d*2 + (kk2 & 1);   // pair-interleaved
            sA[li] = nodeA[g];
            sN[li] = node[g];
        }
        __syncthreads();
        if (k0 + 32 < M_)
            __builtin_prefetch(nodeA + ((size_t)(b*M_ + k0 + 32 + (tid >> 3)))*DG_, 0, 1);
        #pragma unroll
        for (int kk = 0; kk < 8; ++kk) {
            int k = kk*4;
            v2f a1 = *(const v2f*)(gaRow + k0 + k + 2*hlf);
            v2f a2 = *(const v2f*)(spRow + k0 + k + 2*hlf);
            int p = (k >> 1) + hlf;          // k-pair index for this half-wave
            #pragma unroll
            for (int t = 0; t < 5; ++t) {
                int off2 = p*160 + (t*16 + ln)*2;
                v2f b1 = *(const v2f*)(sA + off2);
                v2f b2 = *(const v2f*)(sN + off2);
                acc[t] = __builtin_amdgcn_wmma_f32_16x16x4_f32(false, a1, false, b1,
                                                               (short)0, acc[t], false, false);
                acc[t] = __builtin_amdgcn_wmma_f32_16x16x4_f32(false, a2, false, b2,
                                                               (short)0, acc[t], false, false);
            }
        }
    }
    #pragma unroll
    for (int t = 0; t < 5; ++t) {
        int col = t*16 + ln;
        float tv = tvec[b*DG_ + col];
        #pragma unroll
        for (int r = 0; r < 8; ++r) {
            int row = i0 + r + 8*hlf;
            float v = invdeg[b*M_ + row] * (acc[t][r] + tv);
            agg[((size_t)(b*M_ + row))*DG_ + col] = v;
        }
    }
}

// ---------------------------------------------------------------------------
// 9) WMMA f32: node_out = elu(agg @ gnn_w[l] + gnn_b[l]); weights in LDS
//    (same k-pair-interleaved LDS layout for direct b64 operand loads)
// ---------------------------------------------------------------------------
__global__ void __launch_bounds__(256) k_gnnmm(
    const float* __restrict__ agg, const float* __restrict__ gw,
    const float* __restrict__ gb, float* __restrict__ nodeout)
{
    __shared__ float sW[80*80];
    __shared__ float sB[80];
    int tid = threadIdx.x;
    for (int i = tid; i < 80*80; i += 256) {
        int k = i / 80, d = i % 80;
        sW[(k >> 1)*160 + d*2 + (k & 1)] = gw[i];
    }
    if (tid < 80) sB[tid] = gb[tid];
    __syncthreads();
    int lane = tid & 31, wave = tid >> 5;
    int hlf  = lane >> 4, ln = lane & 15;
    int r0g  = blockIdx.x*128 + wave*16;     // row tile in [0, B*M)
    const float* aRow = agg + (size_t)(r0g + ln)*DG_;
    v8f acc[5] = {};
    #pragma unroll
    for (int k = 0; k < DG_; k += 4) {
        v2f a = *(const v2f*)(aRow + k + 2*hlf);
        int p = (k >> 1) + hlf;
        #pragma unroll
        for (int t = 0; t < 5; ++t) {
            v2f bb = *(const v2f*)(sW + p*160 + (t*16 + ln)*2);
            acc[t] = __builtin_amdgcn_wmma_f32_16x16x4_f32(false, a, false, bb,
                                                           (short)0, acc[t], false, false);
        }
    }
    #pragma unroll
    for (int t = 0; t < 5; ++t) {
        int col = t*16 + ln;
        float bias = sB[col];
        #pragma unroll
        for (int r = 0; r < 8; ++r) {
            int row = r0g + r + 8*hlf;
            float v = acc[t][r] + bias;
            nodeout[(size_t)row*DG_ + col] = v > 0.f ? v : (__expf(v) - 1.f);
        }
    }
}

// ---------------------------------------------------------------------------
// 10) output: res = [node0|node1|node2|hT] @ out_w + out_b
// ---------------------------------------------------------------------------
__global__ void k_out(const float* __restrict__ n0, const float* __restrict__ n1,
                      const float* __restrict__ n2, const float* __restrict__ hT,
                      const float* __restrict__ ow, const float* __restrict__ ob,
                      float* __restrict__ res)
{
    __shared__ float so[272];
    int tid = threadIdx.x;
    for (int i = tid; i < 272; i += 256) so[i] = ow[i];
    __syncthreads();
    int gid = blockIdx.x*256 + tid;
    float a = ob[0];
    const float* p0 = n0 + (size_t)gid*DG_;
    const float* p1 = n1 + (size_t)gid*DG_;
    const float* p2 = n2 + (size_t)gid*DG_;
    #pragma unroll
    for (int d = 0; d < DG_; ++d) a += p0[d]*so[d] + p1[d]*so[80+d] + p2[d]*so[160+d];
    const float* ph = hT + (size_t)gid*32;
    #pragma unroll
    for (int h2 = 0; h2 < 32; ++h2) a += ph[h2]*so[240+h2];
    res[gid] = a;
}

// ---------------------------------------------------------------------------
extern "C" void kernel_launch(void* const* d_in, const int* in_sizes, int n_in,
                              void* d_out, int out_size, void* d_ws, size_t ws_size,
                              hipStream_t stream)
{
    const float* x      = (const float*)d_in[0];
    const float* Wt     = (const float*)d_in[1];
    const float* bt     = (const float*)d_in[2];
    const float* WQw    = (const float*)d_in[3];
    const float* WQb    = (const float*)d_in[4];
    const float* WKw    = (const float*)d_in[5];
    const float* WKb    = (const float*)d_in[6];
    const float* tencw  = (const float*)d_in[7];
    const float* tencb  = (const float*)d_in[8];
    const float* sencw  = (const float*)d_in[9];
    const float* sencb  = (const float*)d_in[10];
    const float* gwih   = (const float*)d_in[11];
    const float* gwhh   = (const float*)d_in[12];
    const float* gbih   = (const float*)d_in[13];
    const float* gbhh   = (const float*)d_in[14];
    const float* Vv     = (const float*)d_in[15];
    const float* bv     = (const float*)d_in[16];
    const float* W1     = (const float*)d_in[17];
    const float* b1     = (const float*)d_in[18];
    const float* W2     = (const float*)d_in[19];
    const float* Wb     = (const float*)d_in[20];
    const float* wb     = (const float*)d_in[21];
    const float* dgate  = (const float*)d_in[22];
    const float* adjg   = (const float*)d_in[23];
    const float* adjo   = (const float*)d_in[24];
    const float* degree = (const float*)d_in[25];
    const float* gnnw   = (const float*)d_in[26];
    const float* gnnb   = (const float*)d_in[27];
    const float* outw   = (const float*)d_in[28];
    const float* outb   = (const float*)d_in[29];
    float* res = (float*)d_out;
    (void)in_sizes; (void)n_in; (void)out_size; (void)ws_size;

    float* ws = (float*)d_ws;
    size_t off = 0;
    float* hsc   = ws + off; off += (size_t)B_*M_*HSC_;
    float* qbuf  = ws + off; off += (size_t)B_*M_*HID_;
    float* kbuf  = ws + off; off += (size_t)B_*M_*HID_;
    float* ksum  = ws + off; off += (size_t)B_*32;
    float* G     = ws + off; off += (size_t)B_*1024;
    float* hT    = ws + off; off += (size_t)B_*M_*NH_;
    float* ann   = ws + off; off += (size_t)B_*M_;
    float* cg    = ws + off; off += (size_t)B_*M_;
    float* colw  = ws + off; off += (size_t)B_*M_;
    float* invd  = ws + off; off += (size_t)B_*M_;
    float* tvec  = ws + off; off += (size_t)B_*DG_;
    float* part  = ws + off; off += (size_t)16*B_*M_;
    float* spat  = ws + off; off += (size_t)M_*M_;
    float* node0 = ws + off; off += (size_t)B_*M_*DG_;
    float* node1 = ws + off; off += (size_t)B_*M_*DG_;
    float* node2 = ws + off; off += (size_t)B_*M_*DG_;
    float* nodeA = ws + off; off += (size_t)B_*M_*DG_;
    float* agg   = ws + off; off += (size_t)B_*M_*DG_;

    dim3 blk256(256);
    k_timeconv<<<dim3(B_*M_/256), blk256, 0, stream>>>(x, Wt, bt, WQw, WQb, WKw, WKb,
                                                       hsc, qbuf, kbuf);
    k_gram<<<dim3(B_), dim3(1024), 0, stream>>>(kbuf, G, ksum);
    k_node0<<<dim3(B_*M_/256), blk256, 0, stream>>>(hsc, qbuf, G, ksum, tencw, tencb,
                                                    sencw, sencb, degree, node0);
    k_gru<<<dim3(B_*M_/256), blk256, 0, stream>>>(x, gwih, gwhh, gbih, gbhh,
                                                  W1, b1, W2, Vv, bv, hT, ann);
    k_cgate_part<<<dim3(M_/128, 16), dim3(128), 0, stream>>>(ann, Wb, part);
    k_cgate_fin<<<dim3(B_*M_/256), blk256, 0, stream>>>(part, ann, wb, cg, colw);
    k_spatial<<<dim3((unsigned)((size_t)M_*M_/256)), blk256, 0, stream>>>(dgate, adjo, degree, spat);
    k_deg<<<dim3(B_*M_), blk256, 0, stream>>>(adjg, spat, cg, colw, invd);

    const float* nin = node0;
    float* nouts[2] = { node1, node2 };
    for (int l = 0; l < 2; ++l) {
        k_nodeA<<<dim3((unsigned)((size_t)B_*M_*DG_/256)), blk256, 0, stream>>>(nin, cg, nodeA);
        k_tvec<<<dim3(B_), dim3(DG_), 0, stream>>>(nin, colw, tvec);
        k_aggmm<<<dim3(M_/128, B_), blk256, 0, stream>>>(adjg, spat, nin, nodeA, tvec, invd, agg);
        k_gnnmm<<<dim3(B_*M_/128), blk256, 0, stream>>>(agg, gnnw + l*DG_*DG_, gnnb + l*DG_, nouts[l]);
        nin = nouts[l];
    }
    k_out<<<dim3(B_*M_/256), blk256, 0, stream>>>(node0, node1, node2, hT, outw, outb, res);
}